// GAT_13683765805594
// MI455X (gfx1250) — compile-verified
//
#include <hip/hip_runtime.h>
#include <math.h>

typedef float v2f __attribute__((ext_vector_type(2)));
typedef float v8f __attribute__((ext_vector_type(8)));

#define LRELU(x, s) ((x) > 0.0f ? (x) : (s) * (x))

// Ordered-int trick for float atomic max (handles mixed signs, -inf init).
__device__ __forceinline__ void atomicMaxF(float* addr, float v) {
  if (v >= 0.0f)
    atomicMax((int*)addr, __float_as_int(v));
  else
    atomicMin((unsigned int*)addr, __float_as_uint(v));
}

// ---------------------------------------------------------------------------
// FP32 WMMA GEMM: D[M,N] = A[M,K] @ B[K,N], one wave per 16x16 tile,
// K stepped by 4 via V_WMMA_F32_16X16X4_F32.  Requires M%16==0, N%16==0, K%4==0
// (true for every layer here: M=10000, K in {128,2560,448}, N in {2560,448,128,16}).
// A-operand layout (ISA 7.12.2, 32-bit A 16x4): lanes 0-15 hold K=k,k+1 of row
// M=lane; lanes 16-31 hold K=k+2,k+3 of row M=lane-16. B mirrors (K striped the
// same way, N across lanes). C/D: VGPR r = row r + 8*(lane>=16), col = lane&15.
// ---------------------------------------------------------------------------
__global__ __launch_bounds__(128) void gat_wmma_gemm(
    const float* __restrict__ A, const float* __restrict__ B,
    float* __restrict__ D, int M, int K, int Nout) {
  const int tileM = blockIdx.x;
  const int tileN = blockIdx.y * blockDim.y + threadIdx.y;  // uniform per wave
  if (tileN * 16 >= Nout) return;  // whole-wave exit: EXEC stays all-ones for WMMA
  const int lane = threadIdx.x;
  const int half = lane >> 4;   // 0: K pair (k,k+1), 1: (k+2,k+3)
  const int l    = lane & 15;
  const size_t arow = (size_t)(tileM * 16 + l);
  const int    bcol = tileN * 16 + l;
  const float* __restrict__ ap = A + arow * (size_t)K;

  v8f acc = {};
  for (int k = 0; k < K; k += 4) {
    const int kk = k + 2 * half;
    v2f a = *(const v2f*)(ap + kk);             // contiguous, 8B-aligned
    v2f b;
    b.x = B[(size_t)kk * Nout + bcol];
    b.y = B[(size_t)(kk + 1) * Nout + bcol];
    acc = __builtin_amdgcn_wmma_f32_16x16x4_f32(
        /*neg_a=*/false, a, /*neg_b=*/false, b,
        /*c_mod=*/(short)0, acc, /*reuse_a=*/false, /*reuse_b=*/false);
  }
#pragma unroll
  for (int r = 0; r < 8; ++r)
    D[(size_t)(tileM * 16 + r + 8 * half) * Nout + tileN * 16 + l] = acc[r];
}

// ---------------------------------------------------------------------------
__global__ void gat_fill(float* __restrict__ p, float v, int n) {
  int i = blockIdx.x * blockDim.x + threadIdx.x;
  if (i < n) p[i] = v;
}

// e_src[n,h] = sum_c h[n,h,c]*a_src[h,c];  e_dst likewise.
__global__ void gat_attn_halves(const float* __restrict__ h,
                                const float* __restrict__ asrc,
                                const float* __restrict__ adst,
                                float* __restrict__ esrc, float* __restrict__ edst,
                                int Nn, int H, int C) {
  int idx = blockIdx.x * blockDim.x + threadIdx.x;  // n*H + hh
  if (idx >= Nn * H) return;
  int n = idx / H, hh = idx - n * H;
  const float* hp = h + (size_t)n * H * C + (size_t)hh * C;
  const float* as = asrc + (size_t)hh * C;
  const float* ad = adst + (size_t)hh * C;
  float s = 0.f, d = 0.f;
  for (int c = 0; c < C; ++c) {
    float v = hp[c];
    s = fmaf(v, as[c], s);
    d = fmaf(v, ad[c], d);
  }
  esrc[idx] = s;
  edst[idx] = d;
}

// Pass 1: per-(edge,head) logit, LeakyReLU(0.2), store; segment max over dst.
__global__ void gat_edge_logits(const int* __restrict__ ei, int E, int NE, int H,
                                const float* __restrict__ esrc,
                                const float* __restrict__ edst,
                                float* __restrict__ ebuf, float* __restrict__ mbuf) {
  int idx = blockIdx.x * blockDim.x + threadIdx.x;  // eid*H + hh
  if (idx >= NE * H) return;
  int eid = idx / H, hh = idx - eid * H;
  int s = (eid < E) ? ei[eid]     : (eid - E);  // self-loops appended
  int d = (eid < E) ? ei[E + eid] : (eid - E);
  float e = esrc[s * H + hh] + edst[d * H + hh];
  e = LRELU(e, 0.2f);
  ebuf[idx] = e;
  atomicMaxF(&mbuf[d * H + hh], e);
}

// Pass 2: ex = exp(e - m[dst]); segment sum -> denom.
__global__ void gat_edge_exp(const int* __restrict__ ei, int E, int NE, int H,
                             float* __restrict__ ebuf, const float* __restrict__ mbuf,
                             float* __restrict__ denom) {
  int idx = blockIdx.x * blockDim.x + threadIdx.x;
  if (idx >= NE * H) return;
  int eid = idx / H, hh = idx - eid * H;
  int d = (eid < E) ? ei[E + eid] : (eid - E);
  float m = mbuf[d * H + hh];
  if (!(m >= -3.0e38f && m <= 3.0e38f)) m = 0.0f;  // isfinite guard
  float ex = __expf(ebuf[idx] - m);
  ebuf[idx] = ex;
  atomicAdd(&denom[d * H + hh], ex);
}

// Pass 3: out[dst,h,c] += (ex/denom) * h[src,h,c]. One block per edge.
__global__ void gat_edge_aggregate(const int* __restrict__ ei, int E, int NE,
                                   int H, int C,
                                   const float* __restrict__ h,
                                   const float* __restrict__ ebuf,
                                   const float* __restrict__ denom,
                                   float* __restrict__ out) {
  int eid = blockIdx.x;
  if (eid >= NE) return;
  int s = (eid < E) ? ei[eid]     : (eid - E);
  int d = (eid < E) ? ei[E + eid] : (eid - E);
  int HC = H * C;
  const float* hs = h + (size_t)s * HC;
  float* od = out + (size_t)d * HC;
  for (int j = threadIdx.x; j < HC; j += blockDim.x) {
    int hh = j / C;
    float alpha = ebuf[(size_t)eid * H + hh] / (denom[d * H + hh] + 1e-16f);
    atomicAdd(&od[j], alpha * hs[j]);
  }
}

__global__ void gat_bias_act(float* __restrict__ out, const float* __restrict__ b,
                             int Nn, int HC, float slope) {
  int i = blockIdx.x * blockDim.x + threadIdx.x;
  if (i >= Nn * HC) return;
  float v = out[i] + b[i % HC];
  out[i] = LRELU(v, slope);
}

// Layer 4 tail: +bias, LeakyReLU(0.1), row softmax over 16 classes.
__global__ void gat_final(const float* __restrict__ in, const float* __restrict__ b,
                          float* __restrict__ outp, int Nn, int Cc) {
  int n = blockIdx.x * blockDim.x + threadIdx.x;
  if (n >= Nn) return;
  float vals[16];
  float mx = -3.4e38f;
  for (int j = 0; j < Cc; ++j) {
    float v = in[(size_t)n * Cc + j] + b[j];
    v = LRELU(v, 0.1f);
    vals[j] = v;
    mx = fmaxf(mx, v);
  }
  float s = 0.f;
  for (int j = 0; j < Cc; ++j) { vals[j] = __expf(vals[j] - mx); s += vals[j]; }
  float inv = 1.0f / s;
  for (int j = 0; j < Cc; ++j) outp[(size_t)n * Cc + j] = vals[j] * inv;
}

// ---------------------------------------------------------------------------
static void run_gat_layer(const float* xin, int Fin,
                          const float* W, const float* as_, const float* ad_,
                          int H, int C, const int* ei, int E, int Nn,
                          float* hbuf, float* obuf,
                          float* esrc, float* edst, float* mbuf, float* denom,
                          float* ebuf, hipStream_t stream) {
  const int HC = H * C;
  const int NE = E + Nn;
  // 1) projection h = x @ W  (WMMA fp32)
  dim3 gb(32, 4);
  dim3 gg(Nn / 16, (HC / 16 + 3) / 4);
  gat_wmma_gemm<<<gg, gb, 0, stream>>>(xin, W, hbuf, Nn, Fin, HC);
  // 2) attention halves
  int t1 = Nn * H;
  gat_attn_halves<<<(t1 + 255) / 256, 256, 0, stream>>>(hbuf, as_, ad_, esrc, edst, Nn, H, C);
  // 3) init segment buffers + output accumulator (xin==obuf already consumed by GEMM)
  gat_fill<<<(t1 + 255) / 256, 256, 0, stream>>>(mbuf, -__builtin_inff(), t1);
  gat_fill<<<(t1 + 255) / 256, 256, 0, stream>>>(denom, 0.0f, t1);
  int t2 = Nn * HC;
  gat_fill<<<(t2 + 255) / 256, 256, 0, stream>>>(obuf, 0.0f, t2);
  // 4) segment softmax + weighted scatter
  int t3 = NE * H;
  gat_edge_logits<<<(t3 + 255) / 256, 256, 0, stream>>>(ei, E, NE, H, esrc, edst, ebuf, mbuf);
  gat_edge_exp<<<(t3 + 255) / 256, 256, 0, stream>>>(ei, E, NE, H, ebuf, mbuf, denom);
  gat_edge_aggregate<<<NE, 256, 0, stream>>>(ei, E, NE, H, C, hbuf, ebuf, denom, obuf);
}

extern "C" void kernel_launch(void* const* d_in, const int* in_sizes, int n_in,
                              void* d_out, int out_size, void* d_ws, size_t ws_size,
                              hipStream_t stream) {
  const float* x  = (const float*)d_in[0];
  const int*   ei = (const int*)d_in[1];
  const float* W1 = (const float*)d_in[2];
  const float* as1 = (const float*)d_in[3];
  const float* ad1 = (const float*)d_in[4];
  const float* b1 = (const float*)d_in[5];
  const float* W2 = (const float*)d_in[6];
  const float* as2 = (const float*)d_in[7];
  const float* ad2 = (const float*)d_in[8];
  const float* b2 = (const float*)d_in[9];
  const float* W3 = (const float*)d_in[10];
  const float* as3 = (const float*)d_in[11];
  const float* ad3 = (const float*)d_in[12];
  const float* b3 = (const float*)d_in[13];
  const float* W4 = (const float*)d_in[14];
  const float* as4 = (const float*)d_in[15];
  const float* ad4 = (const float*)d_in[16];
  const float* b4 = (const float*)d_in[17];

  const int Nn = in_sizes[0] / 128;  // 10000
  const int E  = in_sizes[1] / 2;    // 80000
  const int NE = E + Nn;

  // Workspace carve-up (floats).
  float* ws = (float*)d_ws;
  size_t off = 0;
  float* hbuf  = ws + off; off += (size_t)Nn * 2560;  // projected features (max layer)
  float* obuf  = ws + off; off += (size_t)Nn * 2560;  // aggregated output / next input
  float* esrc  = ws + off; off += (size_t)Nn * 10;
  float* edst  = ws + off; off += (size_t)Nn * 10;
  float* mbuf  = ws + off; off += (size_t)Nn * 10;
  float* denom = ws + off; off += (size_t)Nn * 10;
  float* ebuf  = ws + off; off += (size_t)NE * 10;
  (void)ws_size; (void)n_in; (void)out_size;

  // Layer 1: 128 -> 10 heads x 256, concat
  run_gat_layer(x, 128, W1, as1, ad1, 10, 256, ei, E, Nn,
                hbuf, obuf, esrc, edst, mbuf, denom, ebuf, stream);
  gat_bias_act<<<((Nn * 2560) + 255) / 256, 256, 0, stream>>>(obuf, b1, Nn, 2560, 0.1f);

  // Layer 2: 2560 -> 8 x 56, concat
  run_gat_layer(obuf, 2560, W2, as2, ad2, 8, 56, ei, E, Nn,
                hbuf, obuf, esrc, edst, mbuf, denom, ebuf, stream);
  gat_bias_act<<<((Nn * 448) + 255) / 256, 256, 0, stream>>>(obuf, b2, Nn, 448, 0.1f);

  // Layer 3: 448 -> 4 x 32, concat
  run_gat_layer(obuf, 448, W3, as3, ad3, 4, 32, ei, E, Nn,
                hbuf, obuf, esrc, edst, mbuf, denom, ebuf, stream);
  gat_bias_act<<<((Nn * 128) + 255) / 256, 256, 0, stream>>>(obuf, b3, Nn, 128, 0.1f);

  // Layer 4: 128 -> 1 x 16, mean over heads (H=1 => identity); bias+act+softmax fused.
  run_gat_layer(obuf, 128, W4, as4, ad4, 1, 16, ei, E, Nn,
                hbuf, obuf, esrc, edst, mbuf, denom, ebuf, stream);
  gat_final<<<(Nn + 255) / 256, 256, 0, stream>>>(obuf, b4, (float*)d_out, Nn, 16);
}